// ClassicalEncoderDecoder_29231547417018
// MI455X (gfx1250) — compile-verified
//
#include <hip/hip_runtime.h>
#include <hip/hip_bf16.h>

// ---------------------------------------------------------------------------
// ClassicalEncoderDecoder on MI455X (gfx1250)
//
//   lifted_core closed form:  S = r 1^T - 1 r^T is rank-2, so the Cayley
//   solve (I-S)^{-1}(I+S) = I + c0 + c1 r_i + c2 r_j + c3 r_i r_j (Woodbury).
//   core[p][q] = d_p*delta_pq + f[p] + g[p]*r[q]   (rank-2 + diagonal)
//   lifted[i][j] = sum_{o in window} rot90(core, o%4)[i-o][j-o]
//     -> per-diagonal, per-(rot k), class-mod-4 prefix sums => O(N^2) build.
//
//   GEMM chain (8x): fp32 WMMA V_WMMA_F32_16X16X4_F32 (exact fp32 accumulate,
//   no precision loss through the 8-layer chain). LDS-tiled 128x64, BK=16.
// ---------------------------------------------------------------------------

#define Hh   512
#define H1   511
#define NN   1024
#define BB   16384

typedef float v2f __attribute__((ext_vector_type(2)));
typedef float v8f __attribute__((ext_vector_type(8)));

// ---------------------------------------------------------------------------
// Kernel 1: per matrix, compute Woodbury scalars and f[], g[] arrays.
// grid = 8 blocks (one per matrix), block = 512 threads.
// fg layout: fg[m*1024 + t] = f[t], fg[m*1024 + 512 + t] = g[t]
// ---------------------------------------------------------------------------
__global__ __launch_bounds__(512) void k_build_fg(
    const float* __restrict__ enc_rot, const float* __restrict__ enc_diag,
    const float* __restrict__ dec_rot, const float* __restrict__ dec_diag,
    float* __restrict__ fg) {
  const int m = blockIdx.x;   // 0..7
  const int t = threadIdx.x;  // 0..511
  const float* r = (m < 4) ? (enc_rot + m * Hh) : (dec_rot + (m - 4) * Hh);
  const float* d = (m < 4) ? (enc_diag + m * Hh) : (dec_diag + (m - 4) * Hh);

  __shared__ float sA[512];
  __shared__ float sB[512];
  __shared__ float cS[4];

  const float rv = r[t];
  sA[t] = rv;
  sB[t] = rv * rv;
  __syncthreads();
  for (int off = 256; off > 0; off >>= 1) {
    if (t < off) { sA[t] += sA[t + off]; sB[t] += sB[t + off]; }
    __syncthreads();
  }
  if (t == 0) {
    const float a = sA[0], b = sB[0];
    const float D = 1.0f - a * a + 512.0f * b;
    cS[0] = -2.0f * b / D;                                      // c0
    cS[1] = 1.0f + ((1.0f + a) * (1.0f + a) - 512.0f * b) / D;  // c1
    cS[2] = -1.0f - ((1.0f - a) * (1.0f - a) - 512.0f * b) / D; // c2
    cS[3] = -1024.0f / D;                                       // c3 = -2H/D
  }
  __syncthreads();
  const float dv = d[t];
  fg[m * 1024 + t]       = dv * (cS[0] + cS[1] * rv);  // f[t]
  fg[m * 1024 + 512 + t] = dv * (cS[2] + cS[3] * rv);  // g[t]
}

// ---------------------------------------------------------------------------
// Kernel 2: per matrix m, for each (rot-k, diagonal s) build class-mod-4
// prefix sums P[k][si][t] = h(t) + P[k][si][t-4]   (t = i - o along diag).
//   h_k(t) = value of rot90(core,k)[t][t-s] (0 outside the valid range),
//   with the delta(core diagonal) terms folded in.
// 4092 threads per matrix; P is reused across matrices (stream-ordered).
// ---------------------------------------------------------------------------
__global__ void k_build_prefix(
    int m,
    const float* __restrict__ enc_rot, const float* __restrict__ enc_diag,
    const float* __restrict__ dec_rot, const float* __restrict__ dec_diag,
    const float* __restrict__ fg, float* __restrict__ P) {
  const int idx = blockIdx.x * blockDim.x + threadIdx.x;
  if (idx >= 4 * 1023) return;
  const int k  = idx & 3;
  const int si = idx >> 2;     // 0..1022
  const int s  = si - H1;      // -511..511

  const float* r = (m < 4) ? (enc_rot + m * Hh) : (dec_rot + (m - 4) * Hh);
  const float* d = (m < 4) ? (enc_diag + m * Hh) : (dec_diag + (m - 4) * Hh);
  const float* f = fg + m * 1024;
  const float* g = f + 512;

  float run[4] = {0.f, 0.f, 0.f, 0.f};
  float* Pp = P + ((size_t)(k * 1023 + si)) * Hh;

  const int tlo = (s > 0) ? s : 0;
  const int thi = (s < 0) ? (H1 + s) : H1;

  for (int t = 0; t < Hh; ++t) {
    float h = 0.f;
    if (t >= tlo && t <= thi) {
      const int v = t - s;  // both t,v in [0,512)
      if (k == 0) {                       // core[t][v]
        h = f[t] + g[t] * r[v];
        if (s == 0) h += d[t];
      } else if (k == 1) {                // core[v][H1-t]
        h = f[v] + g[v] * r[H1 - t];
        if (2 * t == H1 + s) h += d[v];
      } else if (k == 2) {                // core[H1-t][H1-v]
        h = f[H1 - t] + g[H1 - t] * r[H1 - v];
        if (s == 0) h += d[H1 - t];
      } else {                            // core[H1-v][t]
        h = f[H1 - v] + g[H1 - v] * r[t];
        if (2 * t == H1 + s) h += d[H1 - v];
      }
    }
    run[t & 3] += h;
    Pp[t] = run[t & 3];
  }
}

// ---------------------------------------------------------------------------
// Kernel 3: assemble lifted[i][j] from the prefix sums.
//  window over t = i-o:  a = max(0, s, i-512),  b = min(i, 511, 511+s)
//  rot index k = (i - t) & 3  ->  t must be in class c = (i-k) & 3.
// ---------------------------------------------------------------------------
__global__ void k_build_lifted(const float* __restrict__ P,
                               float* __restrict__ L) {
  const int idx = blockIdx.x * blockDim.x + threadIdx.x;  // 0 .. 1024*1024-1
  const int i = idx >> 10;
  const int j = idx & 1023;
  const int s = i - j;
  float acc = 0.f;
  if (s > -Hh && s < Hh) {
    const int si = s + H1;
    const int a = max(0, max(s, i - Hh));
    const int b = min(i, min(H1, H1 + s));
    if (b >= a) {
#pragma unroll
      for (int k = 0; k < 4; ++k) {
        const int c   = (i - k) & 3;
        const int Thi = b - ((b + 4 - c) & 3);          // largest t<=b, t&3==c
        const int Tlo = a + ((c + 4 - (a & 3)) & 3);    // smallest t>=a, t&3==c
        if (Thi >= Tlo) {
          const float* Pp = P + ((size_t)(k * 1023 + si)) * Hh;
          float v = Pp[Thi];
          if (Tlo >= 4) v -= Pp[Tlo - 4];
          acc += v;
        }
      }
    }
  }
  L[idx] = acc;
}

// ---------------------------------------------------------------------------
// Kernel 4: Out[M,N] = A[M,K] * W[N,K]^T  (M=16384, N=K=1024), fp32 WMMA.
// Block tile 128(M) x 64(N), BK=16 staged in LDS. 8 waves, each 32x32.
// ISA fragment layouts (CDNA5 ISA):
//   A 16x4 : V0 = {K0 | K2}, V1 = {K1 | K3}   (lanes 0-15 | 16-31, M = lane&15)
//   B 4x16 : mirrored (N = lane&15, K split across VGPR x lane-half)
//   C 16x16: VGPR g -> M = g (lanes 0-15) / g+8 (lanes 16-31), N = lane&15
// ---------------------------------------------------------------------------
#define BM 128
#define BN 64
#define BK 16

__global__ __launch_bounds__(256) void k_gemm_xwt(const float* __restrict__ A,
                                                  const float* __restrict__ W,
                                                  float* __restrict__ Out) {
  __shared__ float As[BM][BK + 1];   // stride 17 -> conflict-free column reads
  __shared__ float Bs[BK][BN + 1];   // Bs[k][n] = W[nb+n][kb+k] (transposed)

  const int tid  = threadIdx.x;
  const int mb   = blockIdx.x * BM;
  const int nb   = blockIdx.y * BN;
  const int lane = tid & 31;
  const int w    = tid >> 5;
  const int wm   = (w & 3) * 32;   // wave tile row offset (4 waves in M)
  const int wn   = (w >> 2) * 32;  // wave tile col offset (2 waves in N)
  const int half = lane >> 4;      // 0: lanes 0-15, 1: lanes 16-31
  const int l15  = lane & 15;

  v8f acc[2][2];
#pragma unroll
  for (int fm = 0; fm < 2; ++fm)
#pragma unroll
    for (int fn = 0; fn < 2; ++fn) acc[fm][fn] = v8f{};

  const int am = tid >> 1;        // 0..127 : A row in tile
  const int ac = (tid & 1) * 8;   // 0 / 8  : A col group
  const int wr = tid >> 2;        // 0..63  : W row (output feature) in tile
  const int wk = (tid & 3) * 4;   // 0,4,8,12 : W col group

  for (int kb = 0; kb < 1024; kb += BK) {
    // ---- stage A tile (128 x 16) ----
    const float4 a0 = *(const float4*)(A + (size_t)(mb + am) * 1024 + kb + ac);
    const float4 a1 = *(const float4*)(A + (size_t)(mb + am) * 1024 + kb + ac + 4);
    As[am][ac + 0] = a0.x; As[am][ac + 1] = a0.y;
    As[am][ac + 2] = a0.z; As[am][ac + 3] = a0.w;
    As[am][ac + 4] = a1.x; As[am][ac + 5] = a1.y;
    As[am][ac + 6] = a1.z; As[am][ac + 7] = a1.w;
    // ---- stage W tile transposed (Bs[k][n] = W[n][k]) ----
    const float4 wv = *(const float4*)(W + (size_t)(nb + wr) * 1024 + kb + wk);
    Bs[wk + 0][wr] = wv.x; Bs[wk + 1][wr] = wv.y;
    Bs[wk + 2][wr] = wv.z; Bs[wk + 3][wr] = wv.w;
    __syncthreads();

#pragma unroll
    for (int k4 = 0; k4 < BK; k4 += 4) {
      v2f af[2], bf[2];
#pragma unroll
      for (int fm = 0; fm < 2; ++fm) {
        const int row = wm + fm * 16 + l15;
        af[fm].x = As[row][k4 + half * 2];       // V0: K0 | K2
        af[fm].y = As[row][k4 + half * 2 + 1];   // V1: K1 | K3
      }
#pragma unroll
      for (int fn = 0; fn < 2; ++fn) {
        const int col = wn + fn * 16 + l15;
        bf[fn].x = Bs[k4 + half * 2][col];
        bf[fn].y = Bs[k4 + half * 2 + 1][col];
      }
#pragma unroll
      for (int fm = 0; fm < 2; ++fm)
#pragma unroll
        for (int fn = 0; fn < 2; ++fn)
          acc[fm][fn] = __builtin_amdgcn_wmma_f32_16x16x4_f32(
              false, af[fm], false, bf[fn], (short)0, acc[fm][fn],
              false, false);
    }
    __syncthreads();
  }

  // ---- store C/D fragments ----
#pragma unroll
  for (int fm = 0; fm < 2; ++fm) {
#pragma unroll
    for (int fn = 0; fn < 2; ++fn) {
      const int col = nb + wn + fn * 16 + l15;
#pragma unroll
      for (int g = 0; g < 8; ++g) {
        const int row = mb + wm + fm * 16 + g + half * 8;
        Out[(size_t)row * 1024 + col] = acc[fm][fn][g];
      }
    }
  }
}

// ---------------------------------------------------------------------------
// Host-side orchestration.
// Workspace layout (floats):
//   fg      : 8 * 1024                       =      8192
//   P       : 4 * 1023 * 512                 = 2,095,104   (reused per matrix)
//   lifted  : 8 * 1024 * 1024                = 8,388,608
//   actA    : 16384 * 1024                   = 16,777,216
//   actB    : 16384 * 1024                   = 16,777,216
// total ~ 176 MB.
// d_out: [bottleneck (16384*1024) | out (16384*1024)] fp32.
// ---------------------------------------------------------------------------
extern "C" void kernel_launch(void* const* d_in, const int* in_sizes, int n_in,
                              void* d_out, int out_size, void* d_ws,
                              size_t ws_size, hipStream_t stream) {
  (void)in_sizes; (void)n_in; (void)out_size; (void)ws_size;

  const float* x        = (const float*)d_in[0];
  const float* enc_rot  = (const float*)d_in[1];
  const float* enc_diag = (const float*)d_in[2];
  const float* dec_rot  = (const float*)d_in[3];
  const float* dec_diag = (const float*)d_in[4];

  float* out = (float*)d_out;
  float* ws  = (float*)d_ws;

  float* fg     = ws;
  float* P      = fg + (size_t)8 * 1024;
  float* lifted = P + (size_t)4 * 1023 * Hh;
  float* actA   = lifted + (size_t)8 * NN * NN;
  float* actB   = actA + (size_t)BB * NN;

  float* bneck  = out;
  float* final_ = out + (size_t)BB * NN;

  // 1) Woodbury scalars + f/g arrays for all 8 matrices
  k_build_fg<<<8, 512, 0, stream>>>(enc_rot, enc_diag, dec_rot, dec_diag, fg);

  // 2) Build the 8 lifted matrices (prefix buffer reused; stream-ordered)
  for (int m = 0; m < 8; ++m) {
    k_build_prefix<<<(4 * 1023 + 255) / 256, 256, 0, stream>>>(
        m, enc_rot, enc_diag, dec_rot, dec_diag, fg, P);
    k_build_lifted<<<(NN * NN) / 256, 256, 0, stream>>>(
        P, lifted + (size_t)m * NN * NN);
  }

  // 3) GEMM chain: s <- s @ lifted^T, eight times
  dim3 grid(BB / BM, NN / BN);
  k_gemm_xwt<<<grid, 256, 0, stream>>>(x,     lifted + 0 * (size_t)NN * NN, actA);
  k_gemm_xwt<<<grid, 256, 0, stream>>>(actA,  lifted + 1 * (size_t)NN * NN, actB);
  k_gemm_xwt<<<grid, 256, 0, stream>>>(actB,  lifted + 2 * (size_t)NN * NN, actA);
  k_gemm_xwt<<<grid, 256, 0, stream>>>(actA,  lifted + 3 * (size_t)NN * NN, bneck);
  k_gemm_xwt<<<grid, 256, 0, stream>>>(bneck, lifted + 4 * (size_t)NN * NN, actA);
  k_gemm_xwt<<<grid, 256, 0, stream>>>(actA,  lifted + 5 * (size_t)NN * NN, actB);
  k_gemm_xwt<<<grid, 256, 0, stream>>>(actB,  lifted + 6 * (size_t)NN * NN, actA);
  k_gemm_xwt<<<grid, 256, 0, stream>>>(actA,  lifted + 7 * (size_t)NN * NN, final_);
}